// LengthRegulator_4432406249715
// MI455X (gfx1250) — compile-verified
//
#include <hip/hip_runtime.h>
#include <hip/hip_bf16.h>
#include <stdint.h>

// LengthRegulator for MI455X (gfx1250, wave32).
// Memory-bound: 201 MB of write-once output -> nontemporal b128 stores;
// x gather stays resident in L2; duration row pulled via async global->LDS.

namespace {
constexpr int kB = 32;
constexpr int kS = 256;
constexpr int kD = 384;
constexpr int kD4 = kD / 4;        // 96 float4 per row
constexpr int kMaxLen = 4096;
constexpr int kRowsPerBlock = 32;  // t-rows per block
constexpr int kRowY = 8;           // blockDim.y
}

// Native clang vector type: accepted by __builtin_nontemporal_store and
// lowers to b128 vector memory ops.
typedef float v4f __attribute__((ext_vector_type(4)));

__global__ __launch_bounds__(768, 1)
void length_regulator_kernel(const float* __restrict__ x,
                             const int* __restrict__ duration,
                             float* __restrict__ out,
                             float* __restrict__ mel_len_out) {
    __shared__ int dur_lds[kS];
    __shared__ int cum[kS];
    __shared__ int wavesum[kS / 32];

    const int b    = blockIdx.y;
    const int tile = blockIdx.x;                      // 0..kMaxLen/kRowsPerBlock-1
    const int tid  = threadIdx.y * 96 + threadIdx.x;  // 0..767

    // ---- Stage 1: duration[b, tid] -> LDS via CDNA5 async load (ASYNCcnt) ----
    if (tid < kS) {
        // Low 32 bits of a generic LDS pointer == wave-relative LDS offset
        // (ISA 10.2: flat->LDS mapping truncates to addr[31:0]).
        uint32_t lds_off = (uint32_t)(uintptr_t)(&dur_lds[tid]);
        uint64_t gaddr   = (uint64_t)(uintptr_t)(duration + (size_t)b * kS + tid);
        asm volatile("global_load_async_to_lds_b32 %0, %1, off"
                     :: "v"(lds_off), "v"(gaddr) : "memory");
        asm volatile("s_wait_asynccnt 0" ::: "memory");

        // ---- Stage 2: inclusive prefix scan (wave32 shuffle scan) ----
        int v = dur_lds[tid];
#pragma unroll
        for (int off = 1; off < 32; off <<= 1) {
            int n = __shfl_up(v, off, 32);
            if ((tid & 31) >= off) v += n;
        }
        if ((tid & 31) == 31) wavesum[tid >> 5] = v;
        cum[tid] = v;  // intra-wave partial
    }
    __syncthreads();
    if (tid < kS) {
        const int w = tid >> 5;
        int add = 0;
#pragma unroll
        for (int i = 0; i < (kS / 32) - 1; ++i)
            if (i < w) add += wavesum[i];
        cum[tid] += add;
    }
    __syncthreads();

    const int mel_len = cum[kS - 1];
    if (tile == 0 && tid == 0) mel_len_out[b] = (float)mel_len;

    // ---- Stage 3: expand-copy tile of 32 output rows ----
    const v4f* __restrict__ xrow = (const v4f*)(x + (size_t)b * kS * kD);
    v4f* __restrict__ obase =
        (v4f*)(out + ((size_t)b * kMaxLen + (size_t)tile * kRowsPerBlock) * kD);

    const int c = threadIdx.x;  // float4 column 0..95
#pragma unroll
    for (int k = 0; k < kRowsPerBlock / kRowY; ++k) {
        const int r = k * kRowY + threadIdx.y;          // row within tile
        const int t = tile * kRowsPerBlock + r;         // global frame index

        // searchsorted(cum, t, 'right'): branchless 8-step binary search.
        // pos = min(count of cum[i] <= t, 255), pos in [0,255].
        int pos = 0;
#pragma unroll
        for (int sh = 7; sh >= 0; --sh) {
            const int probe = pos + (1 << sh);
            if (cum[probe - 1] <= t) pos = probe;
        }
        const int idx = (pos > kS - 1) ? (kS - 1) : pos;

        v4f val;
        if (t < mel_len) {
            val = xrow[(size_t)idx * kD4 + c];          // L2-resident gather
        } else {
            val = (v4f){0.f, 0.f, 0.f, 0.f};
        }
        // Write-once 201MB stream: nontemporal b128 store, keep L2 for x.
        __builtin_nontemporal_store(val, &obase[(size_t)r * kD4 + c]);
    }
}

extern "C" void kernel_launch(void* const* d_in, const int* in_sizes, int n_in,
                              void* d_out, int out_size, void* d_ws, size_t ws_size,
                              hipStream_t stream) {
    const float* x        = (const float*)d_in[0];
    const int*   duration = (const int*)d_in[1];
    // d_in[2] = max_len scalar (== 4096, compile-time constant here)

    float* out         = (float*)d_out;
    float* mel_len_out = out + (size_t)kB * kMaxLen * kD;  // tuple tail

    dim3 grid(kMaxLen / kRowsPerBlock, kB);  // (128, 32)
    dim3 block(96, kRowY);                   // 768 threads = 24 wave32
    hipLaunchKernelGGL(length_regulator_kernel, grid, block, 0, stream,
                       x, duration, out, mel_len_out);
}